// Model2_7687991460345
// MI455X (gfx1250) — compile-verified
//
#include <hip/hip_runtime.h>
#include <hip/hip_bf16.h>

#define SEQ_LEN 262144
#define EMB 300
#define HID 10
#define G4 (4*HID)   // 40 gate values per step

typedef __attribute__((ext_vector_type(2))) float v2f;
typedef __attribute__((ext_vector_type(8))) float v8f;

// ---------------------------------------------------------------------------
// Kernel A: x_gates = x @ W_ih^T + (b_ih + b_hh), stored PERMUTED as
// [S][10][4] (hidden-major, gate-minor) so the scan kernel can read one
// float4 per lane per step.
//
// M = 262144 rows, 4 M-tiles of 16 per wave (64 rows/wave) to amortize the
// B-operand fetch; N = 40 (3 tiles of 16, tail masked at store only);
// K = 300 (75 steps of 4) via V_WMMA_F32_16X16X4_F32.
//
// A layout (32-bit A 16x4): lanes 0-15 -> M=lane, VGPR0=K+0, VGPR1=K+1;
//                           lanes 16-31 -> M=lane-16, VGPR0=K+2, VGPR1=K+3.
// B layout symmetric with lane = N. C/D: lane -> N=lane&15, VGPR v -> M=v+8*hi.
//
// NOTE: B columns 40..47 are fetched from a clamped (row 0) address; the
// resulting D elements with N>=40 are garbage but are never stored, and a
// garbage B column cannot contaminate other columns' dot products.
// ---------------------------------------------------------------------------
__global__ __launch_bounds__(128) void lstm_xgates_wmma(
    const float* __restrict__ x,      // [S,300]
    const float* __restrict__ W_ih,   // [40,300]
    const float* __restrict__ b_ih,   // [40]
    const float* __restrict__ b_hh,   // [40]
    float* __restrict__ xg)           // [S,40] permuted: col = (n%10)*4 + n/10
{
    const int lane   = threadIdx.x & 31;
    const int waveId = threadIdx.x >> 5;
    const int hi     = lane >> 4;          // 0 or 1
    const int l16    = lane & 15;
    const long rowBase = ((long)blockIdx.x * 4 + waveId) * 64;   // 4 tiles of 16

    v8f acc[4][3];
    #pragma unroll
    for (int tm = 0; tm < 4; ++tm)
        #pragma unroll
        for (int tn = 0; tn < 3; ++tn)
            acc[tm][tn] = (v8f){};

    const float* arow[4];
    #pragma unroll
    for (int tm = 0; tm < 4; ++tm)
        arow[tm] = x + (rowBase + tm * 16 + l16) * (long)EMB + 2 * hi;

    const int n0 = l16;
    const int n1 = 16 + l16;
    const int n2 = 32 + l16;                     // 32..47; only <40 valid
    const bool n2ok = (n2 < G4);
    const float* bp[3];
    bp[0] = W_ih + (long)n0 * EMB + 2 * hi;
    bp[1] = W_ih + (long)n1 * EMB + 2 * hi;
    bp[2] = W_ih + (long)(n2ok ? n2 : 0) * EMB + 2 * hi;  // clamped, see note

    #pragma unroll 3
    for (int k = 0; k < EMB; k += 4) {
        v2f a[4], b[3];
        #pragma unroll
        for (int tm = 0; tm < 4; ++tm) a[tm] = *(const v2f*)(arow[tm] + k);
        #pragma unroll
        for (int tn = 0; tn < 3; ++tn) b[tn] = *(const v2f*)(bp[tn] + k);

        #pragma unroll
        for (int tm = 0; tm < 4; ++tm) {
            #pragma unroll
            for (int tn = 0; tn < 3; ++tn) {
                acc[tm][tn] = __builtin_amdgcn_wmma_f32_16x16x4_f32(
                    false, a[tm], false, b[tn], (short)0, acc[tm][tn],
                    false, false);
            }
        }
    }

    const float bias0 = b_ih[n0] + b_hh[n0];
    const float bias1 = b_ih[n1] + b_hh[n1];
    const float bias2 = n2ok ? (b_ih[n2] + b_hh[n2]) : 0.f;

    // permuted column index: gate-major n -> hidden-major (n%10)*4 + n/10
    const int c0 = (n0 % HID) * 4 + n0 / HID;
    const int c1 = (n1 % HID) * 4 + n1 / HID;
    const int c2 = n2ok ? ((n2 % HID) * 4 + n2 / HID) : 0;

    #pragma unroll
    for (int tm = 0; tm < 4; ++tm) {
        #pragma unroll
        for (int v = 0; v < 8; ++v) {
            const long m = rowBase + tm * 16 + v + 8 * hi;
            float* orow = xg + m * (long)G4;
            orow[c0] = acc[tm][0][v] + bias0;
            orow[c1] = acc[tm][1][v] + bias1;
            if (n2ok) orow[c2] = acc[tm][2][v] + bias2;
        }
    }
}

// ---------------------------------------------------------------------------
// Kernel B: the inherently serial LSTM scan. One wave32.
// Lane j (j = lane%10) owns gate rows j / 10+j / 20+j / 30+j of W_hh
// (40 weights register-resident per lane). x_gates is stored [S][10][4],
// so lane j reads its 4 gate pre-activations with ONE b128 load per step.
// h[0..9] is replicated per lane; the new h_j is re-broadcast with 10
// ds_bpermute shuffles. x_gates (42 MB) is L2-resident; prefetch ahead.
// ---------------------------------------------------------------------------
__device__ __forceinline__ float fast_sigmoid(float x) {
    return 1.f / (1.f + __expf(-x));
}
__device__ __forceinline__ float fast_tanh(float x) {
    return 2.f / (1.f + __expf(-2.f * x)) - 1.f;
}

__global__ __launch_bounds__(32) void lstm_scan(
    const float* __restrict__ xg,     // [S,40] permuted [S][10][4]
    const float* __restrict__ W_hh,   // [40,10]
    const float* __restrict__ W_dec,  // [2,10]
    const float* __restrict__ b_dec,  // [2]
    float* __restrict__ out)          // [2]
{
    const int lane = threadIdx.x;     // 0..31
    const int j = lane % HID;         // lanes 10..31 duplicate lanes 0..9

    float Wi[HID], Wf[HID], Wg[HID], Wo[HID];
    #pragma unroll
    for (int k = 0; k < HID; ++k) {
        Wi[k] = W_hh[(0 * HID + j) * HID + k];
        Wf[k] = W_hh[(1 * HID + j) * HID + k];
        Wg[k] = W_hh[(2 * HID + j) * HID + k];
        Wo[k] = W_hh[(3 * HID + j) * HID + k];
    }

    float h[HID];
    #pragma unroll
    for (int k = 0; k < HID; ++k) h[k] = 0.f;
    float c = 0.f;

    for (long t = 0; t < SEQ_LEN; ++t) {
        const float* row = xg + t * (long)G4;
        __builtin_prefetch(row + 32 * G4, 0, 0);   // ~32 steps ahead

        // One aligned b128 load per lane; independent of the recurrence.
        const float4 ld = *(const float4*)(row + j * 4);
        float gi = ld.x, gf = ld.y, gg = ld.z, go = ld.w;

        #pragma unroll
        for (int k = 0; k < HID; ++k) {
            const float hk = h[k];
            gi = fmaf(Wi[k], hk, gi);
            gf = fmaf(Wf[k], hk, gf);
            gg = fmaf(Wg[k], hk, gg);
            go = fmaf(Wo[k], hk, go);
        }

        const float ig = fast_sigmoid(gi);
        const float fg = fast_sigmoid(gf);
        const float gv = fast_tanh(gg);
        const float og = fast_sigmoid(go);

        c = fmaf(fg, c, ig * gv);
        const float hj = og * fast_tanh(c);

        #pragma unroll
        for (int k = 0; k < HID; ++k)
            h[k] = __shfl(hj, k, 32);    // ds_bpermute broadcast of lane k
    }

    if (lane == 0) {
        float z0 = b_dec[0];
        float z1 = b_dec[1];
        #pragma unroll
        for (int k = 0; k < HID; ++k) {
            z0 = fmaf(W_dec[0 * HID + k], h[k], z0);
            z1 = fmaf(W_dec[1 * HID + k], h[k], z1);
        }
        const float m   = fmaxf(z0, z1);
        const float lse = m + __logf(__expf(z0 - m) + __expf(z1 - m));
        out[0] = z0 - lse;
        out[1] = z1 - lse;
    }
}

extern "C" void kernel_launch(void* const* d_in, const int* in_sizes, int n_in,
                              void* d_out, int out_size, void* d_ws, size_t ws_size,
                              hipStream_t stream) {
    (void)in_sizes; (void)n_in; (void)out_size; (void)ws_size;
    const float* x     = (const float*)d_in[0];  // [S,300]
    const float* W_ih  = (const float*)d_in[1];  // [40,300]
    const float* W_hh  = (const float*)d_in[2];  // [40,10]
    const float* b_ih  = (const float*)d_in[3];  // [40]
    const float* b_hh  = (const float*)d_in[4];  // [40]
    const float* W_dec = (const float*)d_in[5];  // [2,10]
    const float* b_dec = (const float*)d_in[6];  // [2]
    float* out = (float*)d_out;                  // [2]
    float* xg  = (float*)d_ws;                   // [S,40] scratch (42 MB)

    // 4 waves per block, 4 M-tiles (64 rows) per wave.
    const int blocks = SEQ_LEN / (64 * 4);       // 1024
    lstm_xgates_wmma<<<blocks, 128, 0, stream>>>(x, W_ih, b_ih, b_hh, xg);
    lstm_scan<<<1, 32, 0, stream>>>(xg, W_hh, W_dec, b_dec, out);
}